// OnePoleFilterCell_48000554500212
// MI455X (gfx1250) — compile-verified
//
#include <hip/hip_runtime.h>

// One-pole time-varying IIR y[t] = a[t]*y[t-1] + b[t] as a 3-pass chunked
// affine scan. Memory-bound (AI ~0.19 FLOP/B): optimize the data path.
// Pass 1: per-chunk affine aggregates (async global->LDS staging, wave32 scan)
// Pass 2: per-row serial scan of 235 chunk aggregates -> chunk entry carries
// Pass 3: re-read (L2-resident: 184MB inputs vs 192MB L2), apply, and stream
//         y out with non-temporal stores so outputs don't evict L2 inputs.

namespace {
constexpr int kB      = 32;
constexpr int kT      = 480000;
constexpr int kItems  = 8;
constexpr int kBlock  = 256;                       // 8 waves (wave32)
constexpr int kChunk  = kBlock * kItems;           // 2048 elements
constexpr int kNChunk = (kT + kChunk - 1) / kChunk;// 235 (last chunk = 768)
constexpr int kWaves  = kBlock / 32;
}

typedef int   v4i_t __attribute__((ext_vector_type(4)));
typedef float v4f_t __attribute__((ext_vector_type(4)));
typedef __attribute__((address_space(1))) void  as1_void;
typedef __attribute__((address_space(3))) void  as3_void;
typedef __attribute__((address_space(1))) v4i_t as1_v4i;
typedef __attribute__((address_space(3))) v4i_t as3_v4i;

// ---- CDNA5 async global->LDS copy (16B quantum), with sync fallback -------
__device__ __forceinline__ void asyncCopy16(const void* gsrc, void* ldst) {
#if __has_builtin(__builtin_amdgcn_global_load_async_to_lds_b128)
  __builtin_amdgcn_global_load_async_to_lds_b128(
      (as1_v4i*)(as1_void*)(void*)(uintptr_t)gsrc,
      (as3_v4i*)(as3_void*)ldst, 0, 0);
#else
  *(v4f_t*)ldst = *(const v4f_t*)gsrc;
#endif
}

// Wait for all outstanding async (ASYNCcnt) transfers from this wave.
// Must always emit a wait: LDS data dependence is invisible to the compiler.
__device__ __forceinline__ void asyncWait() {
#if __has_builtin(__builtin_amdgcn_s_wait_asynccnt)
  __builtin_amdgcn_s_wait_asynccnt(0);
#else
  asm volatile("s_wait_asynccnt 0x0" ::: "memory");
#endif
}

// Inclusive wave32 scan of affine pairs: combine(earlier,later) =
// (A_l*A_e, A_l*B_e + B_l). Non-commutative, associative.
__device__ __forceinline__ void waveInclScan(float& A, float& B, int lane) {
#pragma unroll
  for (int d = 1; d < 32; d <<= 1) {
    float pA = __shfl_up(A, d, 32);
    float pB = __shfl_up(B, d, 32);
    if (lane >= d) {
      B = fmaf(A, pB, B);
      A = A * pA;
    }
  }
}

// ---------------- Pass 1: per-chunk affine aggregate -----------------------
__global__ void __launch_bounds__(kBlock)
k_partials(const float* __restrict__ in, const float* __restrict__ prev,
           float* __restrict__ pA, float* __restrict__ pB) {
  __shared__ float lds[kChunk * 3];   // 24 KB: interleaved [s,g,a] per element
  __shared__ float wA[kWaves], wB[kWaves];

  const int blk  = blockIdx.x;
  const int r    = blk / kNChunk;
  const int c    = blk - r * kNChunk;
  const int tid  = threadIdx.x;
  const int lane = tid & 31;
  const int wave = tid >> 5;

  const int t0     = c * kChunk;
  const int nvalid = (kT - t0 < kChunk) ? (kT - t0) : kChunk;  // 2048 or 768
  const int nq     = (nvalid * 12) >> 4;                       // 16B quanta

  const char* gsrc = (const char*)(in + (size_t)3 * ((size_t)r * kT + t0));
  char*       ldst = (char*)lds;
  for (int q = tid; q < nq; q += kBlock)
    asyncCopy16(gsrc + (size_t)q * 16, ldst + (size_t)q * 16);
  asyncWait();
  __syncthreads();

  // Thread-serial affine composition over 8 elements.
  float A = 1.0f, Bv = 0.0f;
  const int e0 = tid * kItems;
  if (e0 < nvalid) {
    const float pv = (c == 0 && tid == 0) ? prev[r] : 0.0f;
#pragma unroll
    for (int i = 0; i < kItems; ++i) {
      const int   e = e0 + i;
      const float s = lds[3 * e + 0];
      const float g = lds[3 * e + 1];
      const float a = lds[3 * e + 2];
      float b = g * s;
      if (c == 0 && e == 0) b = fmaf(a, pv, b);  // fold prev_state into b[0]
      Bv = fmaf(a, Bv, b);
      A *= a;
    }
  }

  waveInclScan(A, Bv, lane);
  if (lane == 31) { wA[wave] = A; wB[wave] = Bv; }
  __syncthreads();

  if (tid == 0) {
    float eA = 1.0f, eB = 0.0f;
#pragma unroll
    for (int w = 0; w < kWaves; ++w) {
      const float aA = wA[w], aB = wB[w];
      eB = fmaf(aA, eB, aB);
      eA *= aA;
    }
    pA[blk] = eA;
    pB[blk] = eB;
  }
}

// ------------- Pass 2: per-row serial scan over chunk aggregates -----------
__global__ void k_carry(const float* __restrict__ pA,
                        const float* __restrict__ pB,
                        float* __restrict__ carry) {
  const int r = threadIdx.x;
  if (r >= kB) return;
  float v = 0.0f;  // prev_state already folded into chunk 0's b
  const size_t base = (size_t)r * kNChunk;
  for (int c = 0; c < kNChunk; ++c) {
    carry[base + c] = v;                       // value entering chunk c
    v = fmaf(pA[base + c], v, pB[base + c]);
  }
}

// ---------------- Pass 3: apply prefixes, emit y ---------------------------
__global__ void __launch_bounds__(kBlock)
k_apply(const float* __restrict__ in, const float* __restrict__ prev,
        const float* __restrict__ carry, float* __restrict__ out) {
  __shared__ float lds[kChunk * 3];
  __shared__ float wA[kWaves], wB[kWaves], weA[kWaves], weB[kWaves];

  const int blk  = blockIdx.x;
  const int r    = blk / kNChunk;
  const int c    = blk - r * kNChunk;
  const int tid  = threadIdx.x;
  const int lane = tid & 31;
  const int wave = tid >> 5;

  const int t0     = c * kChunk;
  const int nvalid = (kT - t0 < kChunk) ? (kT - t0) : kChunk;
  const int nq     = (nvalid * 12) >> 4;

  const char* gsrc = (const char*)(in + (size_t)3 * ((size_t)r * kT + t0));
  char*       ldst = (char*)lds;
  for (int q = tid; q < nq; q += kBlock)
    asyncCopy16(gsrc + (size_t)q * 16, ldst + (size_t)q * 16);
  asyncWait();
  __syncthreads();

  float a_r[kItems], b_r[kItems];
  float A = 1.0f, Bv = 0.0f;
  const int  e0    = tid * kItems;
  const bool valid = e0 < nvalid;
  if (valid) {
    const float pv = (c == 0 && tid == 0) ? prev[r] : 0.0f;
#pragma unroll
    for (int i = 0; i < kItems; ++i) {
      const int   e = e0 + i;
      const float s = lds[3 * e + 0];
      const float g = lds[3 * e + 1];
      const float a = lds[3 * e + 2];
      float b = g * s;
      if (c == 0 && e == 0) b = fmaf(a, pv, b);
      a_r[i] = a;
      b_r[i] = b;
      Bv = fmaf(a, Bv, b);
      A *= a;
    }
  }

  // Inclusive wave scan, then derive lane-exclusive prefix.
  float iA = A, iB = Bv;
  waveInclScan(iA, iB, lane);
  float leA = __shfl_up(iA, 1, 32);
  float leB = __shfl_up(iB, 1, 32);
  if (lane == 0) { leA = 1.0f; leB = 0.0f; }
  if (lane == 31) { wA[wave] = iA; wB[wave] = iB; }
  __syncthreads();

  if (tid == 0) {
    float eA = 1.0f, eB = 0.0f;
#pragma unroll
    for (int w = 0; w < kWaves; ++w) {
      weA[w] = eA;
      weB[w] = eB;
      const float aA = wA[w], aB = wB[w];
      eB = fmaf(aA, eB, aB);
      eA *= aA;
    }
  }
  __syncthreads();

  // Full exclusive prefix for this thread: wave-exclusive then lane-exclusive.
  const float xA = leA * weA[wave];
  const float xB = fmaf(leA, weB[wave], leB);

  if (valid) {
    float y = fmaf(xA, carry[blk], xB);   // value entering this thread's run
    float o[kItems];
#pragma unroll
    for (int i = 0; i < kItems; ++i) {
      y = fmaf(a_r[i], y, b_r[i]);
      o[i] = y;
    }
    // Non-temporal: stream y to HBM without evicting L2-resident inputs.
    v4f_t* op = (v4f_t*)(out + (size_t)r * kT + t0 + e0);
    v4f_t lo = {o[0], o[1], o[2], o[3]};
    v4f_t hi = {o[4], o[5], o[6], o[7]};
    __builtin_nontemporal_store(lo, op + 0);
    __builtin_nontemporal_store(hi, op + 1);
  }
}

extern "C" void kernel_launch(void* const* d_in, const int* in_sizes, int n_in,
                              void* d_out, int out_size, void* d_ws, size_t ws_size,
                              hipStream_t stream) {
  (void)in_sizes; (void)n_in; (void)out_size; (void)ws_size;
  const float* in   = (const float*)d_in[0];   // [B, T, 3] f32
  const float* prev = (const float*)d_in[1];   // [B] f32
  float* out = (float*)d_out;                  // [B, T] f32

  // Workspace: partialA | partialB | carry  (3 * 32*235 floats = ~90 KB)
  float* pA    = (float*)d_ws;
  float* pB    = pA + (size_t)kB * kNChunk;
  float* carry = pB + (size_t)kB * kNChunk;

  const dim3 grid(kB * kNChunk);  // 7520 blocks
  k_partials<<<grid, kBlock, 0, stream>>>(in, prev, pA, pB);
  k_carry<<<1, 32, 0, stream>>>(pA, pB, carry);
  k_apply<<<grid, kBlock, 0, stream>>>(in, prev, carry, out);
}